// GATLayer_35854386987541
// MI455X (gfx1250) — compile-verified
//
#include <hip/hip_runtime.h>
#include <math.h>

#define NN 50000
#define NE 625000
#define D 128
#define H 4
#define DH 32
#define NEG_SLOPE 0.2f
#define LN_EPS 1e-5f

typedef float v2f __attribute__((ext_vector_type(2)));
typedef float v8f __attribute__((ext_vector_type(8)));

// Monotonic unsigned encoding of float for atomicMax-based segment max.
__device__ __forceinline__ unsigned fenc(float x) {
    unsigned b = __float_as_uint(x);
    return (b & 0x80000000u) ? ~b : (b | 0x80000000u);
}
__device__ __forceinline__ float fdec(unsigned u) {
    unsigned b = (u & 0x80000000u) ? (u & 0x7FFFFFFFu) : ~u;
    return __uint_as_float(b);
}
__device__ __forceinline__ float lrelu(float v) {
    return v > 0.0f ? v : NEG_SLOPE * v;
}

// ---------------- init: zero accumulator, m = enc(-inf), den = 0 -------------
__global__ void k_init(float* __restrict__ out, unsigned* __restrict__ m,
                       float* __restrict__ den) {
    int i = blockIdx.x * 256 + threadIdx.x;      // grid == NN*D exactly
    out[i] = 0.0f;
    if (i < NN * H) { m[i] = 0x007FFFFFu; den[i] = 0.0f; }  // enc(-inf)
}

// ---------------- projection GEMM: X[NN,128] = A[NN,128] @ W[128,128] --------
// One wave per 16x16 output tile; f32 WMMA, K step = 4.
__global__ void k_proj(const float* __restrict__ A, const float* __restrict__ W,
                       float* __restrict__ X) {
    int wave = (blockIdx.x * blockDim.x + threadIdx.x) >> 5;  // 25000 waves
    int lane = threadIdx.x & 31;
    int tm = wave >> 3, tn = wave & 7;
    int row = lane & 15, half = lane >> 4;
    const float* arow = A + (tm * 16 + row) * D;
    const float* bcol = W + tn * 16 + row;       // column = lane&15
    v8f c = {0.f, 0.f, 0.f, 0.f, 0.f, 0.f, 0.f, 0.f};
    for (int k0 = 0; k0 < D; k0 += 4) {
        int ka = k0 + half * 2;                  // ISA 16x4 f32 A layout
        v2f a; a[0] = arow[ka];       a[1] = arow[ka + 1];
        v2f b; b[0] = bcol[ka * D];   b[1] = bcol[(ka + 1) * D];
        c = __builtin_amdgcn_wmma_f32_16x16x4_f32(false, a, false, b,
                                                  (short)0, c, false, false);
    }
    float* drow = X + tn * 16 + row;
#pragma unroll
    for (int j = 0; j < 8; ++j)                  // C: VGPR j -> M = j + 8*half
        drow[(tm * 16 + half * 8 + j) * D] = c[j];
}

// ---------------- per-(node,head) attention logits ---------------------------
__global__ void k_logits(const float* __restrict__ X,
                         const float* __restrict__ att_src,
                         const float* __restrict__ att_dst,
                         float* __restrict__ a_src, float* __restrict__ a_dst) {
    int i = blockIdx.x * blockDim.x + threadIdx.x;
    if (i >= NN * H) return;
    int n = i >> 2, h = i & 3;
    const float* xr = X + n * D + h * DH;
    const float* as = att_src + h * DH;
    const float* ad = att_dst + h * DH;
    float s1 = 0.f, s2 = 0.f;
#pragma unroll 8
    for (int j = 0; j < DH; ++j) { float v = xr[j]; s1 += v * as[j]; s2 += v * ad[j]; }
    a_src[i] = s1; a_dst[i] = s2;
}

// ---------------- edge pass 1: segment max ----------------------------------
__global__ void k_edge_max(const int* __restrict__ ei,
                           const float* __restrict__ a_src,
                           const float* __restrict__ a_dst,
                           unsigned* __restrict__ m) {
    int e = blockIdx.x * blockDim.x + threadIdx.x;
    if (e >= NE) return;
    int s = ei[e], d = ei[NE + e];
#pragma unroll
    for (int h = 0; h < H; ++h) {
        float v = lrelu(a_src[s * H + h] + a_dst[d * H + h]);
        atomicMax(&m[d * H + h], fenc(v));
    }
}

// ---------------- edge pass 2: softmax denominator --------------------------
__global__ void k_edge_sum(const int* __restrict__ ei,
                           const float* __restrict__ a_src,
                           const float* __restrict__ a_dst,
                           const unsigned* __restrict__ m,
                           float* __restrict__ den) {
    int e = blockIdx.x * blockDim.x + threadIdx.x;
    if (e >= NE) return;
    int s = ei[e], d = ei[NE + e];
#pragma unroll
    for (int h = 0; h < H; ++h) {
        float v = lrelu(a_src[s * H + h] + a_dst[d * H + h]);
        atomicAdd(&den[d * H + h], __expf(v - fdec(m[d * H + h])));
    }
}

// ---------------- edge pass 3: weighted scatter-add (one wave per edge) -----
__global__ void k_aggr(const int* __restrict__ ei,
                       const float* __restrict__ a_src,
                       const float* __restrict__ a_dst,
                       const unsigned* __restrict__ m,
                       const float* __restrict__ den,
                       const float* __restrict__ X, float* __restrict__ out) {
    int wave = (blockIdx.x * blockDim.x + threadIdx.x) >> 5;  // == edge id
    int lane = threadIdx.x & 31;
    int s = ei[wave], d = ei[NE + wave];
    int h = lane >> 3;                              // features lane*4..+3 share head
    float v = lrelu(a_src[s * H + h] + a_dst[d * H + h]);
    float alpha = __expf(v - fdec(m[d * H + h])) / fmaxf(den[d * H + h], 1e-16f);
    const float4 xv = *(const float4*)(X + s * D + lane * 4);
    float* o = out + d * D + lane * 4;
    atomicAdd(o + 0, alpha * xv.x);
    atomicAdd(o + 1, alpha * xv.y);
    atomicAdd(o + 2, alpha * xv.z);
    atomicAdd(o + 3, alpha * xv.w);
}

// ---------------- LayerNorm: one wave per node ------------------------------
__global__ void k_ln(const float* __restrict__ acc, const float* __restrict__ bias,
                     const float* __restrict__ gamma, const float* __restrict__ beta,
                     float* __restrict__ lnout) {
    int n = (blockIdx.x * blockDim.x + threadIdx.x) >> 5;     // 50000 waves exact
    int lane = threadIdx.x & 31;
    float4 v = *(const float4*)(acc + n * D + lane * 4);
    const float4 b4 = *(const float4*)(bias + lane * 4);
    v.x += b4.x; v.y += b4.y; v.z += b4.z; v.w += b4.w;
    float s = v.x + v.y + v.z + v.w;
#pragma unroll
    for (int off = 16; off >= 1; off >>= 1) s += __shfl_xor(s, off, 32);
    float mu = s * (1.0f / D);
    float dx = v.x - mu, dy = v.y - mu, dz = v.z - mu, dw = v.w - mu;
    float q = dx * dx + dy * dy + dz * dz + dw * dw;
#pragma unroll
    for (int off = 16; off >= 1; off >>= 1) q += __shfl_xor(q, off, 32);
    float rs = rsqrtf(q * (1.0f / D) + LN_EPS);
    const float4 g4 = *(const float4*)(gamma + lane * 4);
    const float4 be4 = *(const float4*)(beta + lane * 4);
    float4 o;
    o.x = dx * rs * g4.x + be4.x;  o.y = dy * rs * g4.y + be4.y;
    o.z = dz * rs * g4.z + be4.z;  o.w = dw * rs * g4.w + be4.w;
    *(float4*)(lnout + n * D + lane * 4) = o;
}

// ---------------- gate GEMM (K=256) + sigmoid + multiply --------------------
__global__ void k_gate(const float* __restrict__ LN, const float* __restrict__ query,
                       const int* __restrict__ n2g, const float* __restrict__ gW,
                       const float* __restrict__ gb, float* __restrict__ out) {
    int wave = (blockIdx.x * blockDim.x + threadIdx.x) >> 5;  // 25000 waves
    int lane = threadIdx.x & 31;
    int tm = wave >> 3, tn = wave & 7;
    int row = lane & 15, half = lane >> 4;
    int arow_i = tm * 16 + row;
    const float* lrow = LN + arow_i * D;
    const float* qrow = query + n2g[arow_i] * D;
    const float* bcol = gW + tn * 16 + row;
    v8f c = {0.f, 0.f, 0.f, 0.f, 0.f, 0.f, 0.f, 0.f};
    for (int k0 = 0; k0 < D; k0 += 4) {          // K = 0..127 from LN-out
        int ka = k0 + half * 2;
        v2f a; a[0] = lrow[ka];     a[1] = lrow[ka + 1];
        v2f b; b[0] = bcol[ka * D]; b[1] = bcol[(ka + 1) * D];
        c = __builtin_amdgcn_wmma_f32_16x16x4_f32(false, a, false, b,
                                                  (short)0, c, false, false);
    }
    for (int k0 = 0; k0 < D; k0 += 4) {          // K = 128..255 from query
        int ka = k0 + half * 2;
        v2f a; a[0] = qrow[ka];             a[1] = qrow[ka + 1];
        v2f b; b[0] = bcol[(D + ka) * D];   b[1] = bcol[(D + ka + 1) * D];
        c = __builtin_amdgcn_wmma_f32_16x16x4_f32(false, a, false, b,
                                                  (short)0, c, false, false);
    }
    int cc = tn * 16 + row;
    float gbv = gb[cc];
#pragma unroll
    for (int j = 0; j < 8; ++j) {
        int rr = tm * 16 + half * 8 + j;
        float gate = 1.0f / (1.0f + __expf(-(c[j] + gbv)));
        out[rr * D + cc] = LN[rr * D + cc] * gate;
    }
}

extern "C" void kernel_launch(void* const* d_in, const int* in_sizes, int n_in,
                              void* d_out, int out_size, void* d_ws, size_t ws_size,
                              hipStream_t stream) {
    (void)in_sizes; (void)n_in; (void)out_size; (void)ws_size;
    const float* node_input = (const float*)d_in[0];
    const int*   edge_index = (const int*)d_in[1];
    const int*   node2graph = (const int*)d_in[2];
    const float* query      = (const float*)d_in[3];
    const float* W          = (const float*)d_in[4];
    const float* att_src    = (const float*)d_in[5];
    const float* att_dst    = (const float*)d_in[6];
    const float* bias       = (const float*)d_in[7];
    const float* ln_gamma   = (const float*)d_in[8];
    const float* ln_beta    = (const float*)d_in[9];
    const float* gate_W     = (const float*)d_in[10];
    const float* gate_b     = (const float*)d_in[11];
    float* out = (float*)d_out;

    float*    X     = (float*)d_ws;              // NN*D floats (reused as LN-out)
    float*    a_src = X + NN * D;                // NN*H
    float*    a_dst = a_src + NN * H;            // NN*H
    unsigned* m     = (unsigned*)(a_dst + NN * H);  // NN*H
    float*    den   = (float*)(m + NN * H);      // NN*H   (~28.8 MB total)

    k_init    <<<(NN * D) / 256, 256, 0, stream>>>(out, m, den);
    k_proj    <<<6250, 128, 0, stream>>>(node_input, W, X);
    k_logits  <<<(NN * H + 255) / 256, 256, 0, stream>>>(X, att_src, att_dst, a_src, a_dst);
    k_edge_max<<<(NE + 255) / 256, 256, 0, stream>>>(edge_index, a_src, a_dst, m);
    k_edge_sum<<<(NE + 255) / 256, 256, 0, stream>>>(edge_index, a_src, a_dst, m, den);
    k_aggr    <<<NE / 8, 256, 0, stream>>>(edge_index, a_src, a_dst, m, den, X, out);
    k_ln      <<<NN / 8, 256, 0, stream>>>(out, bias, ln_gamma, ln_beta, X);
    k_gate    <<<6250, 128, 0, stream>>>(X, query, node2graph, gate_W, gate_b, out);
}